// PointHead_42150809043450
// MI455X (gfx1250) — compile-verified
//
#include <hip/hip_runtime.h>
#include <math.h>

typedef __attribute__((ext_vector_type(2))) float f32x2;
typedef __attribute__((ext_vector_type(8))) float f32x8;

#define BATCH   16
#define KN      80000
#define NP      100
#define BN      95
#define COVN    5
#define NC      3
#define HM      32
#define PIX_M   (HM*HM)
#define HF      128
#define PIX_F   (HF*HF)
#define CFEAT   512
#define K1      515
#define K1P     516          // layer-1 K padded; 516 mod 64 == 4 -> conflict-free LDS rows
#define HID     256
#define HIDL    260          // hidden K row stride; 260 mod 64 == 4 -> conflict-free
#define NPAD    112          // N=100 padded to 7 WMMA tiles
#define NTILES  7

// ---- workspace layout (float offsets) ----
#define WS_T1      0                          // B*1024
#define WS_T2      (WS_T1 + BATCH*PIX_M)      // B*1024
#define WS_UNC     (WS_T2 + BATCH*PIX_M)      // B*80000
#define WS_TOPK    (WS_UNC + BATCH*KN)        // B*95 ints (rounded to 1536)
#define WS_DMI     (WS_TOPK + 1536)           // descM idx: B*100*4 ints
#define WS_DMW     (WS_DMI + BATCH*NP*4)      // descM w
#define WS_DFI     (WS_DMW + BATCH*NP*4)      // descF idx
#define WS_DFW     (WS_DFI + BATCH*NP*4)      // descF w
#define WS_FEAT    (WS_DFW + BATCH*NP*4)      // featT: B*112*516
#define WS_H1      (WS_FEAT + BATCH*NPAD*K1P) // h1T:   B*112*260
#define WS_H2      (WS_H1 + BATCH*NPAD*HIDL)  // h2T:   B*112*260
#define WS_W1P     (WS_H2 + BATCH*NPAD*HIDL)  // 256*516 padded w1
#define WS_W4P     (WS_W1P + HID*K1P)         // 16*256  padded w4

// ---- output layout (float offsets) ----
#define OUT_REND   0                          // 16*3*100
#define OUT_PTS    (BATCH*NC*NP)              // 16*100*2
#define OUT_MASK   (OUT_PTS + BATCH*NP*2)     // 16*3*1024

// Kernel 1: per-pixel softmax of 3-class mask, keep top-2 sorted probs; copy mask out.
__global__ void k_softmax_sort(const float* __restrict__ mask, float* __restrict__ t1,
                               float* __restrict__ t2, float* __restrict__ out_mask) {
    int g = blockIdx.x * blockDim.x + threadIdx.x;       // B*1024 threads
    int b = g >> 10, p = g & 1023;
    float m0 = mask[(b*NC + 0)*PIX_M + p];
    float m1 = mask[(b*NC + 1)*PIX_M + p];
    float m2 = mask[(b*NC + 2)*PIX_M + p];
    float mx = fmaxf(m0, fmaxf(m1, m2));
    float e0 = expf(m0 - mx), e1 = expf(m1 - mx), e2 = expf(m2 - mx);
    float inv = 1.0f / (e0 + e1 + e2);
    float p0 = e0*inv, p1 = e1*inv, p2 = e2*inv;
    float hi = fmaxf(p0, fmaxf(p1, p2));
    float lo = fminf(p0, fminf(p1, p2));
    t1[g] = hi;
    t2[g] = p0 + p1 + p2 - hi - lo;                      // middle value
    out_mask[(b*NC + 0)*PIX_M + p] = m0;
    out_mask[(b*NC + 1)*PIX_M + p] = m1;
    out_mask[(b*NC + 2)*PIX_M + p] = m2;
}

// Kernel 2: bilinear-sample top1/top2 maps (LDS resident) at 80000 pts/batch.
__global__ void k_uncert(const float* __restrict__ og, const float* __restrict__ t1,
                         const float* __restrict__ t2, float* __restrict__ unc) {
    __shared__ float s1[PIX_M], s2[PIX_M];
    int b = blockIdx.y;
    for (int i = threadIdx.x; i < PIX_M; i += blockDim.x) {
        s1[i] = t1[b*PIX_M + i];
        s2[i] = t2[b*PIX_M + i];
    }
    __syncthreads();
    int i = blockIdx.x * blockDim.x + threadIdx.x;
    if (i >= KN) return;
    float x = og[((size_t)b*KN + i)*2 + 0];
    float y = og[((size_t)b*KN + i)*2 + 1];
    float ix = x*HM - 0.5f, iy = y*HM - 0.5f;
    float x0f = floorf(ix), y0f = floorf(iy);
    float fx = ix - x0f, fy = iy - y0f;
    int x0 = (int)x0f, y0 = (int)y0f;
    float a1 = 0.f, a2 = 0.f;
    #pragma unroll
    for (int c = 0; c < 4; ++c) {
        int xc = x0 + (c & 1), yc = y0 + (c >> 1);
        float w = ((c & 1) ? fx : 1.f - fx) * ((c >> 1) ? fy : 1.f - fy);
        bool v = (xc >= 0) & (xc < HM) & (yc >= 0) & (yc < HM);
        int xi = min(max(xc, 0), HM-1), yi = min(max(yc, 0), HM-1);
        float wv = v ? w : 0.f;
        a1 += wv * s1[yi*HM + xi];
        a2 += wv * s2[yi*HM + xi];
    }
    unc[(size_t)b*KN + i] = a2 - a1;                     // -(top1 - top2)
}

// Kernel 3: top-95 by iterative masked argmax; tie-break lowest index (lax.top_k order).
__global__ void k_topk(float* __restrict__ unc, int* __restrict__ topk) {
    __shared__ float sv[1024];
    __shared__ int   si[1024];
    int b = blockIdx.x, tid = threadIdx.x;
    float* u = unc + (size_t)b*KN;
    for (int it = 0; it < BN; ++it) {
        float bv = -INFINITY; int bi = 0x7fffffff;
        for (int j = tid; j < KN; j += 1024) {
            float v = u[j];
            if (v > bv || (v == bv && j < bi)) { bv = v; bi = j; }
        }
        sv[tid] = bv; si[tid] = bi;
        __syncthreads();
        for (int s = 512; s > 0; s >>= 1) {
            if (tid < s) {
                float ov = sv[tid+s]; int oi = si[tid+s];
                if (ov > sv[tid] || (ov == sv[tid] && oi < si[tid])) { sv[tid] = ov; si[tid] = oi; }
            }
            __syncthreads();
        }
        if (tid == 0) { topk[b*BN + it] = si[0]; u[si[0]] = -INFINITY; }
        __threadfence();
        __syncthreads();
    }
}

__device__ inline void mk_desc(float x, float y, int S, int* di, float* dw) {
    float ix = x*S - 0.5f, iy = y*S - 0.5f;
    float x0f = floorf(ix), y0f = floorf(iy);
    float fx = ix - x0f, fy = iy - y0f;
    int x0 = (int)x0f, y0 = (int)y0f;
    #pragma unroll
    for (int c = 0; c < 4; ++c) {
        int xc = x0 + (c & 1), yc = y0 + (c >> 1);
        float w = ((c & 1) ? fx : 1.f - fx) * ((c >> 1) ? fy : 1.f - fy);
        bool v = (xc >= 0) & (xc < S) & (yc >= 0) & (yc < S);
        int xi = min(max(xc, 0), S-1), yi = min(max(yc, 0), S-1);
        di[c] = yi*S + xi;
        dw[c] = v ? w : 0.f;
    }
}

// Kernel 4: gather points (importance + coverage), emit points output + bilinear descriptors.
__global__ void k_points(const float* __restrict__ og, const float* __restrict__ cov,
                         const int* __restrict__ topk, float* __restrict__ out_pts,
                         int* __restrict__ dmi, float* __restrict__ dmw,
                         int* __restrict__ dfi, float* __restrict__ dfw) {
    int b = blockIdx.x, i = threadIdx.x;
    if (i >= NP) return;
    float x, y;
    if (i < BN) {
        int j = topk[b*BN + i];
        x = og[((size_t)b*KN + j)*2 + 0];
        y = og[((size_t)b*KN + j)*2 + 1];
    } else {
        x = cov[(b*COVN + (i-BN))*2 + 0];
        y = cov[(b*COVN + (i-BN))*2 + 1];
    }
    out_pts[(b*NP + i)*2 + 0] = x;
    out_pts[(b*NP + i)*2 + 1] = y;
    mk_desc(x, y, HM, dmi + (b*NP+i)*4, dmw + (b*NP+i)*4);
    mk_desc(x, y, HF, dfi + (b*NP+i)*4, dfw + (b*NP+i)*4);
}

// Kernel 5: build TRANSPOSED padded feature matrix featT[b][112][516]:
// featT[b][n][k] = bilinear sample of channel k at point n (k<515, n<100), else 0.
__global__ void k_feat(const float* __restrict__ mask, const float* __restrict__ feature,
                       const int* __restrict__ dmi, const float* __restrict__ dmw,
                       const int* __restrict__ dfi, const float* __restrict__ dfw,
                       float* __restrict__ featT) {
    int g = blockIdx.x * blockDim.x + threadIdx.x;       // B*112*516 threads exactly
    int k = g % K1P;  int t = g / K1P;
    int n = t % NPAD; int b = t / NPAD;
    float val = 0.f;
    if (n < NP && k < K1) {
        const int*   di = (k < NC) ? dmi + (b*NP+n)*4 : dfi + (b*NP+n)*4;
        const float* dw = (k < NC) ? dmw + (b*NP+n)*4 : dfw + (b*NP+n)*4;
        const float* src = (k < NC) ? mask + ((size_t)b*NC + k)*PIX_M
                                    : feature + ((size_t)b*CFEAT + (k-NC))*PIX_F;
        val = dw[0]*src[di[0]] + dw[1]*src[di[1]] + dw[2]*src[di[2]] + dw[3]*src[di[3]];
    }
    featT[(size_t)b*NPAD*K1P + n*K1P + k] = val;
}

// Kernel 5b: zero-pad weights into [Mpad, Kpad] so the GEMM A-stream is branch-free.
__global__ void k_padw(const float* __restrict__ W, float* __restrict__ Wp,
                       int Mreal, int Kreal, int Kpad, int total) {
    int g = blockIdx.x * blockDim.x + threadIdx.x;
    if (g >= total) return;
    int k = g % Kpad, m = g / Kpad;
    Wp[g] = (m < Mreal && k < Kreal) ? W[m*Kreal + k] : 0.f;
}

// Kernel 6: fp32 WMMA GEMM, B panel transposed [N][ldsK] so each WMMA B-operand is
// one contiguous ds_load_b64. Workgroup = one batch x 4 M-tiles; whole B^T panel is
// async-staged into LDS (global layout == LDS layout, linear b128 stream).
// A layout (16x4 f32): lanes 0-15 hold K=k0,k0+1; lanes 16-31 hold K=k0+2,k0+3.
// C/D layout: VGPR r <-> row mt*16 + r + 8*(lane>=16), col nt*16 + (lane&15).
__global__ void k_wmma_gemm(const float* __restrict__ Wp, int Kpad, int ldsK,
                            const float* __restrict__ Bmat, int bStrideB,
                            float* __restrict__ Cmat, int bStrideC, int ldC,
                            const float* __restrict__ bias, int mode,
                            int Mtiles, int MgroupsPerBatch) {
    extern __shared__ float sB[];                        // only LDS in kernel -> offset 0
    int b  = blockIdx.x / MgroupsPerBatch;
    int mg = blockIdx.x % MgroupsPerBatch;
    int wave = threadIdx.x >> 5;
    int lane = threadIdx.x & 31;

    // ---- async-stage B^T panel (NPAD x ldsK f32, contiguous) into LDS ----
    unsigned long long gbase =
        (unsigned long long)(Bmat + (size_t)b * bStrideB);
    int chunks = (NPAD * ldsK) >> 2;                     // 16-byte chunks
    for (int c = threadIdx.x; c < chunks; c += blockDim.x) {
        unsigned off = ((unsigned)c) << 4;               // LDS byte addr == global byte off
        asm volatile("global_load_async_to_lds_b128 %0, %1, %2"
                     :: "v"(off), "v"(off), "s"(gbase)
                     : "memory");
    }
    asm volatile("s_wait_asynccnt 0x0" ::: "memory");
    __syncthreads();

    int mt = mg * 4 + wave;
    if (mt < Mtiles) {                                   // wave-uniform; EXEC stays all-ones
        int half = lane >> 4;
        int l    = lane & 15;
        int mA   = mt*16 + l;
        const f32x2* Wp2 = (const f32x2*)Wp;
        f32x8 acc[NTILES] = {};
        for (int k0 = 0; k0 < Kpad; k0 += 4) {
            int ka = k0 + half*2;
            f32x2 a = Wp2[(mA*Kpad + ka) >> 1];          // unconditional b64 load
            f32x2 bv[NTILES];
            #pragma unroll
            for (int nt = 0; nt < NTILES; ++nt)          // 7 contiguous ds_load_b64
                bv[nt] = *(const f32x2*)&sB[(nt*16 + l)*ldsK + ka];
            #pragma unroll
            for (int nt = 0; nt < NTILES; ++nt)          // 7 back-to-back WMMAs
                acc[nt] = __builtin_amdgcn_wmma_f32_16x16x4_f32(
                    false, a, false, bv[nt], (short)0, acc[nt], false, false);
        }
        float* Cb = Cmat + (size_t)b * bStrideC;
        #pragma unroll
        for (int nt = 0; nt < NTILES; ++nt) {
            int n = nt*16 + l;
            if (mode == 0) {
                // transposed store -> next layer's B^T panel; 8 consecutive floats/lane
                #pragma unroll
                for (int r = 0; r < 8; ++r) {
                    int m = mt*16 + r + half*8;
                    float v = acc[nt][r];
                    Cb[n*ldC + m] = v > 0.f ? v : 0.f;   // ReLU
                }
            } else {
                #pragma unroll
                for (int r = 0; r < 8; ++r) {
                    int m = mt*16 + r + half*8;
                    float v = acc[nt][r];
                    if (m < NC && n < NP) Cb[m*NP + n] = v + bias[m]; // +bias -> rend
                }
            }
        }
    }
}

extern "C" void kernel_launch(void* const* d_in, const int* in_sizes, int n_in,
                              void* d_out, int out_size, void* d_ws, size_t ws_size,
                              hipStream_t stream) {
    (void)in_sizes; (void)n_in; (void)out_size; (void)ws_size;
    const float* feature = (const float*)d_in[1];
    const float* mask    = (const float*)d_in[2];
    const float* og      = (const float*)d_in[3];
    const float* cov     = (const float*)d_in[4];
    const float* w1      = (const float*)d_in[5];
    const float* w2      = (const float*)d_in[6];
    const float* w3      = (const float*)d_in[7];
    const float* w4      = (const float*)d_in[8];
    const float* b4      = (const float*)d_in[9];

    float* ws  = (float*)d_ws;
    float* out = (float*)d_out;

    float* t1    = ws + WS_T1;
    float* t2    = ws + WS_T2;
    float* unc   = ws + WS_UNC;
    int*   topk  = (int*)(ws + WS_TOPK);
    int*   dmi   = (int*)(ws + WS_DMI);
    float* dmw   = ws + WS_DMW;
    int*   dfi   = (int*)(ws + WS_DFI);
    float* dfw   = ws + WS_DFW;
    float* featT = ws + WS_FEAT;
    float* h1    = ws + WS_H1;
    float* h2    = ws + WS_H2;
    float* w1p   = ws + WS_W1P;
    float* w4p   = ws + WS_W4P;

    float* out_rend = out + OUT_REND;
    float* out_pts  = out + OUT_PTS;
    float* out_mask = out + OUT_MASK;

    // 1) softmax + sort + mask passthrough
    k_softmax_sort<<<(BATCH*PIX_M)/256, 256, 0, stream>>>(mask, t1, t2, out_mask);
    // 2) uncertainty at 80000 pts/batch
    k_uncert<<<dim3((KN + 255)/256, BATCH), 256, 0, stream>>>(og, t1, t2, unc);
    // 3) top-95 per batch
    k_topk<<<BATCH, 1024, 0, stream>>>(unc, topk);
    // 4) points + bilinear descriptors
    k_points<<<BATCH, 128, 0, stream>>>(og, cov, topk, out_pts, dmi, dmw, dfi, dfw);
    // 5) transposed feature matrix featT[B,112,516] + padded weights
    k_feat<<<(BATCH*NPAD*K1P)/256, 256, 0, stream>>>(mask, feature, dmi, dmw, dfi, dfw, featT);
    k_padw<<<(HID*K1P + 255)/256, 256, 0, stream>>>(w1, w1p, HID, K1, K1P, HID*K1P);
    k_padw<<<(16*HID + 255)/256, 256, 0, stream>>>(w4, w4p, NC, HID, HID, 16*HID);

    // 6) MLP via fp32 WMMA (async-LDS staged transposed B panels): 515->256->256->256->3
    k_wmma_gemm<<<BATCH*4, 128, NPAD*K1P*4, stream>>>(w1p, K1P, K1P, featT, NPAD*K1P,
                                                      h1, NPAD*HIDL, HIDL, nullptr, 0, 16, 4);
    k_wmma_gemm<<<BATCH*4, 128, NPAD*HIDL*4, stream>>>(w2, HID, HIDL, h1, NPAD*HIDL,
                                                       h2, NPAD*HIDL, HIDL, nullptr, 0, 16, 4);
    k_wmma_gemm<<<BATCH*4, 128, NPAD*HIDL*4, stream>>>(w3, HID, HIDL, h2, NPAD*HIDL,
                                                       h1, NPAD*HIDL, HIDL, nullptr, 0, 16, 4);
    k_wmma_gemm<<<BATCH,   128, NPAD*HIDL*4, stream>>>(w4p, HID, HIDL, h1, NPAD*HIDL,
                                                       out_rend, 0, 0, b4, 1, 1, 1);
}